// AdultConnectomeNetwork_75831942578756
// MI455X (gfx1250) — compile-verified
//
#include <hip/hip_runtime.h>

// Problem constants (match reference).
#define NDIM 2048
#define NNZC (NDIM * 64)

constexpr int BM = 128;       // block tile rows
constexpr int BN = 64;        // block tile cols
constexpr int BK = 16;        // k-stage depth
constexpr int LDA_S = BK + 4; // 20: b64 frag loads from 16 rows hit distinct banks
constexpr int LDB_S = BN + 8; // 72: 2*72 % 64 == 16 -> lane halves on disjoint banks
constexpr int STEPS = NDIM / BK; // 128

typedef float v2f __attribute__((ext_vector_type(2)));
typedef float v4f __attribute__((ext_vector_type(4)));
typedef float v8f __attribute__((ext_vector_type(8)));

// Types matching the async-copy builtin signature:
//   void __builtin_amdgcn_global_load_async_to_lds_b128(int4 AS1*, int4 AS3*, imm, imm)
typedef int vi4 __attribute__((vector_size(16)));
typedef __attribute__((address_space(1))) vi4 as1_vi4;
typedef __attribute__((address_space(3))) vi4 as3_vi4;

// ---- CDNA5 async global->LDS copy (ASYNCcnt-tracked), with safe fallback ----
__device__ __forceinline__ void async_copy16(const float* gp, float* lp) {
#if __has_builtin(__builtin_amdgcn_global_load_async_to_lds_b128)
    __builtin_amdgcn_global_load_async_to_lds_b128((as1_vi4*)gp, (as3_vi4*)lp, 0, 0);
#else
    *(v4f*)lp = *(const v4f*)gp; // synchronous fallback
#endif
}

__device__ __forceinline__ void async_wait_all() {
#if __has_builtin(__builtin_amdgcn_s_wait_asynccnt)
    __builtin_amdgcn_s_wait_asynccnt(0);
#elif __has_builtin(__builtin_amdgcn_global_load_async_to_lds_b128)
    asm volatile("s_wait_asynccnt 0x0" ::: "memory");
#endif
}

// ---------------------------------------------------------------- zero ws
__global__ __launch_bounds__(256) void zero_f32x4(v4f* __restrict__ p, int n4) {
    int i = blockIdx.x * blockDim.x + threadIdx.x;
    if (i < n4) {
        v4f z = {0.f, 0.f, 0.f, 0.f};
        p[i] = z;
    }
}

// ------------------------------------------------- COO scatter (dup add)
__global__ __launch_bounds__(256) void scatter_coo(const int* __restrict__ rows,
                                                   const int* __restrict__ cols,
                                                   const float* __restrict__ av,
                                                   const float* __restrict__ wv,
                                                   float* __restrict__ A,
                                                   float* __restrict__ W) {
    int i = blockIdx.x * blockDim.x + threadIdx.x;
    if (i < NNZC) {
        size_t off = (size_t)rows[i] * NDIM + cols[i];
        atomicAdd(&A[off], av[i]);
        atomicAdd(&W[off], wv[i]);
    }
}

// ------------------------------------------------------- f32 WMMA GEMM
// C = A(2048x2048) * B(2048x2048) [+ bias per column]
// grid = (N/BN, N/BM), block = 256 (8 waves). Wave (wm,wn) owns a 32x32 tile
// = 2x2 tiles of 16x16 via V_WMMA_F32_16X16X4_F32. Stages are double-buffered
// in LDS and filled with GLOBAL_LOAD_ASYNC_TO_LDS_B128 so the copy for stage
// k+1 runs underneath the 16 WMMAs of stage k.
__global__ __launch_bounds__(256) void gemm_f32_wmma(const float* __restrict__ A,
                                                     const float* __restrict__ B,
                                                     const float* __restrict__ bias,
                                                     float* __restrict__ C) {
    __shared__ float As[2][BM][LDA_S]; // row-major MxK stage
    __shared__ float Bs[2][BK][LDB_S]; // row-major KxN stage

    const int tid   = threadIdx.x;
    const int lane  = tid & 31;
    const int wave  = tid >> 5;
    const int wm    = wave & 3;   // 0..3 -> 32-row group
    const int wn    = wave >> 2;  // 0..1 -> 32-col group
    const int lmod  = lane & 15;  // M (A) / N (B,C) index within 16
    const int lhalf = lane >> 4;  // selects K-pair (A/B) or M-half (C)

    const int rowBase = blockIdx.y * BM;
    const int colBase = blockIdx.x * BN;

    v8f acc00 = {0.f,0.f,0.f,0.f,0.f,0.f,0.f,0.f};
    v8f acc01 = {0.f,0.f,0.f,0.f,0.f,0.f,0.f,0.f};
    v8f acc10 = {0.f,0.f,0.f,0.f,0.f,0.f,0.f,0.f};
    v8f acc11 = {0.f,0.f,0.f,0.f,0.f,0.f,0.f,0.f};

    // Per-thread chunk coordinates (16B chunks) for the stage copies.
    const int aRow0 = tid >> 2;                 // chunk 0: rows 0..63
    const int aRow1 = (tid + 256) >> 2;         // chunk 1: rows 64..127
    const int aCol  = (tid & 3) * 4;            // 0,4,8,12
    const int bRow  = tid >> 4;                 // 0..15
    const int bCol  = (tid & 15) * 4;           // 0..60

    auto issueStage = [&](int kk, int buf) {
        const int k0 = kk * BK;
        async_copy16(A + (size_t)(rowBase + aRow0) * NDIM + k0 + aCol,
                     &As[buf][aRow0][aCol]);
        async_copy16(A + (size_t)(rowBase + aRow1) * NDIM + k0 + aCol,
                     &As[buf][aRow1][aCol]);
        async_copy16(B + (size_t)(k0 + bRow) * NDIM + colBase + bCol,
                     &Bs[buf][bRow][bCol]);
    };

    auto computeStage = [&](int buf) {
        // Batch ALL fragment loads for the stage, then burst 16 WMMAs.
        v2f aF[4][2];
        v2f bF[4][2];
#pragma unroll
        for (int ks = 0; ks < BK / 4; ++ks) {
            const int kb = ks * 4 + lhalf * 2;  // K-pair base for this lane-half
            aF[ks][0] = *(const v2f*)&As[buf][wm * 32 + lmod][kb];
            aF[ks][1] = *(const v2f*)&As[buf][wm * 32 + 16 + lmod][kb];
            bF[ks][0].x = Bs[buf][kb][wn * 32 + lmod];
            bF[ks][0].y = Bs[buf][kb + 1][wn * 32 + lmod];
            bF[ks][1].x = Bs[buf][kb][wn * 32 + 16 + lmod];
            bF[ks][1].y = Bs[buf][kb + 1][wn * 32 + 16 + lmod];
        }
#pragma unroll
        for (int ks = 0; ks < BK / 4; ++ks) {
            // (neg_a, A, neg_b, B, c_mod, C, reuse_a, reuse_b)
            acc00 = __builtin_amdgcn_wmma_f32_16x16x4_f32(false, aF[ks][0], false, bF[ks][0],
                                                          (short)0, acc00, false, false);
            acc01 = __builtin_amdgcn_wmma_f32_16x16x4_f32(false, aF[ks][0], false, bF[ks][1],
                                                          (short)0, acc01, false, false);
            acc10 = __builtin_amdgcn_wmma_f32_16x16x4_f32(false, aF[ks][1], false, bF[ks][0],
                                                          (short)0, acc10, false, false);
            acc11 = __builtin_amdgcn_wmma_f32_16x16x4_f32(false, aF[ks][1], false, bF[ks][1],
                                                          (short)0, acc11, false, false);
        }
    };

    // Software pipeline: stage kk+1 copies fly while stage kk computes.
    issueStage(0, 0);
    async_wait_all();
    __syncthreads();
    for (int kk = 0; kk < STEPS; ++kk) {
        const int buf = kk & 1;
        if (kk + 1 < STEPS) issueStage(kk + 1, buf ^ 1); // buf^1 readers done @ kk-1 barrier
        computeStage(buf);
        if (kk + 1 < STEPS) {
            async_wait_all();   // our copies have landed in LDS
            __syncthreads();    // everyone's landed; everyone done reading buf
        }
    }

    // Epilogue: C/D layout -> VGPR r is row (r + 8*lhalf), col lmod.
    const int r0 = rowBase + wm * 32 + lhalf * 8;
    const int c0 = colBase + wn * 32 + lmod;
    float bv0 = 0.f, bv1 = 0.f;
    if (bias) { bv0 = bias[c0]; bv1 = bias[c0 + 16]; }
#pragma unroll
    for (int r = 0; r < 8; ++r) {
        C[(size_t)(r0 + r) * NDIM + c0]           = acc00[r] + bv0;
        C[(size_t)(r0 + r) * NDIM + c0 + 16]      = acc01[r] + bv1;
        C[(size_t)(r0 + 16 + r) * NDIM + c0]      = acc10[r] + bv0;
        C[(size_t)(r0 + 16 + r) * NDIM + c0 + 16] = acc11[r] + bv1;
    }
}

// ---------------------------------------------------------------- launch
extern "C" void kernel_launch(void* const* d_in, const int* in_sizes, int n_in,
                              void* d_out, int out_size, void* d_ws, size_t ws_size,
                              hipStream_t stream) {
    const float* x    = (const float*)d_in[0];
    const int*   rows = (const int*)d_in[1];
    const int*   cols = (const int*)d_in[2];
    const float* adjv = (const float*)d_in[3];
    const float* wv   = (const float*)d_in[4];
    const float* bias = (const float*)d_in[5];
    float* out = (float*)d_out;

    const size_t NN = (size_t)NDIM * NDIM;
    float* Amat = (float*)d_ws;        // 16 MB
    float* Wmat = Amat + NN;           // 16 MB
    float* Mmat = Amat + 2 * NN;       // 16 MB
    // After M is built, A/W slots are dead -> reuse as x ping-pong buffers.
    float* xb0 = Amat;
    float* xb1 = Wmat;

    // 1) Zero dense A|W (contiguous 2*NN floats).
    {
        int n4 = (int)((2 * NN) / 4);
        zero_f32x4<<<dim3((n4 + 255) / 256), dim3(256), 0, stream>>>((v4f*)Amat, n4);
    }
    // 2) Scatter COO (duplicates sum -> atomicAdd).
    scatter_coo<<<dim3((NNZC + 255) / 256), dim3(256), 0, stream>>>(rows, cols, adjv, wv,
                                                                    Amat, Wmat);
    // 3) M = A @ W
    dim3 grid(NDIM / BN, NDIM / BM);
    dim3 block(256);
    gemm_f32_wmma<<<grid, block, 0, stream>>>(Amat, Wmat, nullptr, Mmat);
    // 4) Four layers: x = M @ x + bias
    gemm_f32_wmma<<<grid, block, 0, stream>>>(Mmat, x,   bias, xb0);
    gemm_f32_wmma<<<grid, block, 0, stream>>>(Mmat, xb0, bias, xb1);
    gemm_f32_wmma<<<grid, block, 0, stream>>>(Mmat, xb1, bias, xb0);
    gemm_f32_wmma<<<grid, block, 0, stream>>>(Mmat, xb0, bias, out);
}